// QLSTM_65481071406424
// MI455X (gfx1250) — compile-verified
//
#include <hip/hip_runtime.h>

typedef float v2f __attribute__((ext_vector_type(2)));
typedef float v4f __attribute__((ext_vector_type(4)));
typedef float v8f __attribute__((ext_vector_type(8)));

#define SEQ_LEN   128
#define BATCH     4096
#define INPUT_DIM 64
#define DIN       68
#define ROWS      (SEQ_LEN * BATCH)   // 524288 rows of the hoisted GEMM

// ---------------------------------------------------------------------------
// Kernel 1: preX[row][go] = sum_{d<64} X[row][d] * W[go][d] + bias[go]
// using V_WMMA_F32_16X16X4_F32 (16 k-steps of K=4 per 16x16 output tile).
// row = t*BATCH + b ; go = gate*4 + wire.
// A loads are double-buffered across tiles so the 16 loads of tile t+1 are
// in flight while tile t's 16 WMMAs execute (avoids load->wmma serialization).
// ---------------------------------------------------------------------------
__global__ __launch_bounds__(256) void qlstm_gemm_wmma(
    const float* __restrict__ X,
    const float* __restrict__ Wf, const float* __restrict__ Wi,
    const float* __restrict__ Wu, const float* __restrict__ Wo,
    const float* __restrict__ bf, const float* __restrict__ bi,
    const float* __restrict__ bu, const float* __restrict__ bo,
    float* __restrict__ preX)
{
    const int lane = threadIdx.x & 31;
    const int wave = threadIdx.x >> 5;
    const int n    = lane & 15;   // output column go = g*4 + o
    const int half = lane >> 4;   // 0: lanes 0-15, 1: lanes 16-31
    const int g = n >> 2;
    const int o = n & 3;

    const float* Wg = (g == 0) ? Wf : (g == 1) ? Wi : (g == 2) ? Wu : Wo;
    const float* bg = (g == 0) ? bf : (g == 1) ? bi : (g == 2) ? bu : bo;

    // B operand (constant across tiles): per k-step kk, lane holds
    // B[4*kk + 2*half + {0,1}][n] = Wg[o*DIN + 4*kk + 2*half + {0,1}]
    v2f bk[16];
#pragma unroll
    for (int kk = 0; kk < 16; ++kk) {
        const int kbase = 4 * kk + 2 * half;
        bk[kk] = *(const v2f*)(Wg + o * DIN + kbase);
    }
    const float bias = bg[o];

    const int TPW   = 4;                              // tiles per wave
    const int tile0 = (blockIdx.x * 8 + wave) * TPW;  // 32768 tiles total

    // double-buffered A operands: 16 distinct v2f regs per buffer so all 16
    // loads of a tile can be issued as one clause with a single wait.
    v2f abuf[2][16];
    {
        const float* arow =
            X + (size_t)(tile0 * 16 + n) * INPUT_DIM + 2 * half;
#pragma unroll
        for (int kk = 0; kk < 16; ++kk)
            abuf[0][kk] = *(const v2f*)(arow + 4 * kk);
    }

#pragma unroll
    for (int tt = 0; tt < TPW; ++tt) {
        const int cb = tt & 1;        // compile-time after full unroll
        const int nb = cb ^ 1;

        if (tt + 1 < TPW) {           // issue next tile's loads first
            const float* arow =
                X + (size_t)((tile0 + tt + 1) * 16 + n) * INPUT_DIM + 2 * half;
#pragma unroll
            for (int kk = 0; kk < 16; ++kk)
                abuf[nb][kk] = *(const v2f*)(arow + 4 * kk);
        }

        v8f acc;
#pragma unroll
        for (int v = 0; v < 8; ++v) acc[v] = bias;   // bias depends on N only

#pragma unroll
        for (int kk = 0; kk < 16; ++kk) {
            acc = __builtin_amdgcn_wmma_f32_16x16x4_f32(
                /*neg_a=*/false, abuf[cb][kk], /*neg_b=*/false, bk[kk],
                /*c_mod=*/(short)0, acc, /*reuse_a=*/false, /*reuse_b=*/false);
        }

        const int R = (tile0 + tt) * 16;
        float* orow = preX + (size_t)R * 16;
#pragma unroll
        for (int v = 0; v < 8; ++v)
            orow[(size_t)(v + 8 * half) * 16 + n] = acc[v];
    }
}

// ---------------------------------------------------------------------------
// Kernel 2: sequential scan. One lane per (batch, gate); lane g also owns
// hidden index j=g and its cx[j]. LDS exchanges z (16 floats/batch) and hx
// (4 floats/batch) with two workgroup barriers per step.
// ---------------------------------------------------------------------------
__device__ __forceinline__ float fast_sigmoid(float x) {
    return 1.0f / (1.0f + __expf(-x));
}
__device__ __forceinline__ float fast_tanh(float x) {
    // 1 - 2/(e^{2x}+1): saturates cleanly to +/-1, no inf/inf NaN.
    return 1.0f - 2.0f / (__expf(2.0f * x) + 1.0f);
}

__global__ __launch_bounds__(256) void qlstm_scan(
    const float* __restrict__ preX,
    const float* __restrict__ Wf, const float* __restrict__ Wi,
    const float* __restrict__ Wu, const float* __restrict__ Wo,
    const float* __restrict__ rxf, const float* __restrict__ rxi,
    const float* __restrict__ rxu, const float* __restrict__ rxo,
    float* __restrict__ out)
{
    const int tid = threadIdx.x;
    const int g   = tid & 3;        // gate id == owned hidden index
    const int bl  = tid >> 2;       // 0..63 local batch
    const int b   = blockIdx.x * 64 + bl;

    const float* Wg  = (g == 0) ? Wf  : (g == 1) ? Wi  : (g == 2) ? Wu  : Wo;
    const float* rxg = (g == 0) ? rxf : (g == 1) ? rxi : (g == 2) ? rxu : rxo;

    // recurrent weight slice W[g*4+o][64+j] and trainable RX angles (uniform
    // across lanes of same g -> mostly scalarized by the compiler)
    float whx[4][4];
#pragma unroll
    for (int o = 0; o < 4; ++o)
#pragma unroll
        for (int j = 0; j < 4; ++j)
            whx[o][j] = Wg[o * DIN + 64 + j];
    float rx[4];
#pragma unroll
    for (int o = 0; o < 4; ++o) rx[o] = rxg[o];

    __shared__ v4f zsh[64][4];   // [local batch][gate] -> z[0..3]
    __shared__ v4f hxsh[64];     // [local batch]       -> hx[0..3]

    v4f  hx = {0.f, 0.f, 0.f, 0.f};
    float cx = 0.f;
    float hxlast = 0.f;

    const float* prebase = preX + ((size_t)b * 16) + g * 4;
    const size_t tstride = (size_t)BATCH * 16;

    v4f cur = *(const v4f*)(prebase);   // t = 0

    for (int t = 0; t < SEQ_LEN; ++t) {
        // double-buffer next step's preX row (addresses are recurrence-free)
        const int tn = (t + 1 < SEQ_LEN) ? (t + 1) : t;
        v4f nxt = *(const v4f*)(prebase + (size_t)tn * tstride);
        const int tp = (t + 8 < SEQ_LEN) ? (t + 8) : (SEQ_LEN - 1);
        __builtin_prefetch(prebase + (size_t)tp * tstride, 0, 0);

        // pre-activation angles for this gate; product-state probabilities
        float c2[4], s2[4];
#pragma unroll
        for (int o = 0; o < 4; ++o) {
            float pre = cur[o] + whx[o][0] * hx[0] + whx[o][1] * hx[1]
                               + whx[o][2] * hx[2] + whx[o][3] * hx[3];
            float th = (pre + rx[o]) * 0.5f;    // RX(x)RX(theta) = RX(x+theta)
            float c = __cosf(th), s = __sinf(th);
            c2[o] = c * c;
            s2[o] = s * s;
        }

        // Z expectations: CNOT chain is a compile-time bit permutation.
        v4f z = {0.f, 0.f, 0.f, 0.f};
#pragma unroll
        for (int idx = 0; idx < 16; ++idx) {
            const int b0 = (idx >> 3) & 1, b1 = (idx >> 2) & 1;
            const int b2 = (idx >> 1) & 1, b3 = idx & 1;
            float p = (b0 ? s2[0] : c2[0]) * (b1 ? s2[1] : c2[1])
                    * (b2 ? s2[2] : c2[2]) * (b3 ? s2[3] : c2[3]);
            const int f1 = b1 ^ b0;
            const int f2 = b2 ^ f1;
            const int f3 = b3 ^ f2;
            const int f0 = b0 ^ f3;
            z[0] += f0 ? -p : p;
            z[1] += f1 ? -p : p;
            z[2] += f2 ? -p : p;
            z[3] += f3 ? -p : p;
        }

        zsh[bl][g] = z;
        __syncthreads();

        // lane g updates hidden index j = g from the 4 gates' measurements
        const float* zc = (const float*)&zsh[bl][0];
        const float fg = fast_sigmoid(zc[0 * 4 + g]);
        const float ig = fast_sigmoid(zc[1 * 4 + g]);
        const float gg = fast_tanh  (zc[2 * 4 + g]);
        const float og = fast_sigmoid(zc[3 * 4 + g]);

        cx = fg * cx + ig * gg;
        const float h = og * fast_tanh(cx);

        ((float*)&hxsh[bl])[g] = h;
        out[((size_t)t * BATCH + b) * 4 + g] = h;
        hxlast = h;
        __syncthreads();

        hx  = hxsh[bl];
        cur = nxt;
    }

    // trailing (hx, cx) outputs
    const size_t base = (size_t)SEQ_LEN * BATCH * 4;
    out[base + (size_t)b * 4 + g]                       = hxlast;
    out[base + (size_t)BATCH * 4 + (size_t)b * 4 + g]   = cx;
}

// ---------------------------------------------------------------------------
extern "C" void kernel_launch(void* const* d_in, const int* in_sizes, int n_in,
                              void* d_out, int out_size, void* d_ws, size_t ws_size,
                              hipStream_t stream) {
    (void)in_sizes; (void)n_in; (void)out_size; (void)ws_size;
    const float* inputs = (const float*)d_in[0];
    const float* Wf = (const float*)d_in[1];
    const float* bf = (const float*)d_in[2];
    const float* Wi = (const float*)d_in[3];
    const float* bi = (const float*)d_in[4];
    const float* Wu = (const float*)d_in[5];
    const float* bu = (const float*)d_in[6];
    const float* Wo = (const float*)d_in[7];
    const float* bo = (const float*)d_in[8];
    const float* rxf = (const float*)d_in[9];
    const float* rxi = (const float*)d_in[10];
    const float* rxu = (const float*)d_in[11];
    const float* rxo = (const float*)d_in[12];
    float* out  = (float*)d_out;
    float* preX = (float*)d_ws;   // 524288 * 16 floats = 32 MB scratch

    // 32768 tiles of 16 rows; 8 waves/block * 4 tiles/wave -> 1024 blocks
    qlstm_gemm_wmma<<<dim3(ROWS / 16 / 8 / 4), dim3(256), 0, stream>>>(
        inputs, Wf, Wi, Wu, Wo, bf, bi, bu, bo, preX);

    // 64 batch elements * 4 gates per 256-thread block -> 64 blocks
    qlstm_scan<<<dim3(BATCH / 64), dim3(256), 0, stream>>>(
        preX, Wf, Wi, Wu, Wo, rxf, rxi, rxu, rxo, out);
}